// OccupancyManager_23570780521245
// MI455X (gfx1250) — compile-verified
//
#include <hip/hip_runtime.h>
#include <cstdint>

// Instant-NGP style multiresolution hash-grid encoding for MI455X (gfx1250).
// Gather-bound: 67M random float2 gathers from a 64MB table that fits in the
// 192MB L2. Keep the table RT-cached in L2; stream xyz/out with NT hints.
// All 128 gather addresses depend only on xyz, so the backend scheduler can
// hoist loads freely across the fully unrolled level loop (verified: it
// builds per-level SGPR bases and clauses the b64 gathers).

#define L_LEVELS 16
#define T_SIZE   (1u << 19)
#define T_MASK   (T_SIZE - 1u)
#define PRIME1   2654435761u
#define PRIME2   805459861u

typedef float v4f __attribute__((ext_vector_type(4)));

// res[l] = 16 * (2^(1/3))^l  (B = exp(ln(512/16)/15) = 2^(1/3))
__constant__ float c_res[L_LEVELS] = {
    16.0f,               20.15873679831797f,  25.398416831491197f, 32.0f,
    40.31747359663594f,  50.796833662982394f, 64.0f,               80.63494719327188f,
    101.59366732596479f, 128.0f,              161.26989438654377f, 203.18733465192958f,
    256.0f,              322.53978877308754f, 406.37466930385915f, 512.0f
};

__global__ __launch_bounds__(256)
void hashgrid_encode_kernel(const float* __restrict__ xyz,
                            const float2* __restrict__ table,
                            float* __restrict__ out,
                            int n)
{
    const int i = blockIdx.x * blockDim.x + threadIdx.x;
    if (i >= n) return;

    // xyz is read exactly once -> NT load (preserve L2 for the 64MB table).
    const float x = __builtin_nontemporal_load(xyz + 3 * (size_t)i + 0);
    const float y = __builtin_nontemporal_load(xyz + 3 * (size_t)i + 1);
    const float z = __builtin_nontemporal_load(xyz + 3 * (size_t)i + 2);

    float acc[2 * L_LEVELS];

#pragma unroll
    for (int l = 0; l < L_LEVELS; ++l) {
        const float r  = c_res[l];
        const float px = x * r, py = y * r, pz = z * r;
        const float bx = floorf(px), by = floorf(py), bz = floorf(pz);
        const float fx = px - bx,   fy = py - by,   fz = pz - bz;

        // base corner integer coords (xyz in [0,1) -> non-negative)
        const uint32_t ix = (uint32_t)(int)bx;
        const uint32_t iy = (uint32_t)(int)by;
        const uint32_t iz = (uint32_t)(int)bz;

        // Incremental prime hash: 2 integer muls per level total.
        const uint32_t hx0 = ix;                  // * PRIME0 (=1)
        const uint32_t hx1 = ix + 1u;
        const uint32_t hy0 = iy * PRIME1, hy1 = hy0 + PRIME1;
        const uint32_t hz0 = iz * PRIME2, hz1 = hz0 + PRIME2;

        const uint32_t h00 = hx0 ^ hy0;
        const uint32_t h10 = hx1 ^ hy0;
        const uint32_t h01 = hx0 ^ hy1;
        const uint32_t h11 = hx1 ^ hy1;

        const float2* __restrict__ tl = table + (size_t)l * T_SIZE;

        // 8 independent b64 gathers; T = 2^19 -> "% T" is a single AND.
        const float2 e0 = tl[(h00 ^ hz0) & T_MASK];   // (0,0,0)
        const float2 e1 = tl[(h10 ^ hz0) & T_MASK];   // (1,0,0)
        const float2 e2 = tl[(h01 ^ hz0) & T_MASK];   // (0,1,0)
        const float2 e3 = tl[(h11 ^ hz0) & T_MASK];   // (1,1,0)
        const float2 e4 = tl[(h00 ^ hz1) & T_MASK];   // (0,0,1)
        const float2 e5 = tl[(h10 ^ hz1) & T_MASK];   // (1,0,1)
        const float2 e6 = tl[(h01 ^ hz1) & T_MASK];   // (0,1,1)
        const float2 e7 = tl[(h11 ^ hz1) & T_MASK];   // (1,1,1)

        const float gx = 1.0f - fx, gy = 1.0f - fy, gz = 1.0f - fz;
        const float w00 = gx * gy, w10 = fx * gy, w01 = gx * fy, w11 = fx * fy;

        // Four independent FMA chains (depth 4) instead of two (depth 8):
        // more co-issue opportunity for the dual VALU while gathers drain.
        float a0 = 0.0f, a1 = 0.0f, b0 = 0.0f, b1 = 0.0f, w;
        w = w00 * gz; a0 = fmaf(w, e0.x, a0); a1 = fmaf(w, e0.y, a1);
        w = w10 * gz; a0 = fmaf(w, e1.x, a0); a1 = fmaf(w, e1.y, a1);
        w = w01 * gz; a0 = fmaf(w, e2.x, a0); a1 = fmaf(w, e2.y, a1);
        w = w11 * gz; a0 = fmaf(w, e3.x, a0); a1 = fmaf(w, e3.y, a1);
        w = w00 * fz; b0 = fmaf(w, e4.x, b0); b1 = fmaf(w, e4.y, b1);
        w = w10 * fz; b0 = fmaf(w, e5.x, b0); b1 = fmaf(w, e5.y, b1);
        w = w01 * fz; b0 = fmaf(w, e6.x, b0); b1 = fmaf(w, e6.y, b1);
        w = w11 * fz; b0 = fmaf(w, e7.x, b0); b1 = fmaf(w, e7.y, b1);

        acc[2 * l + 0] = a0 + b0;
        acc[2 * l + 1] = a1 + b1;
    }

    // 128 contiguous bytes per lane, streamed with NT stores so the 64MB
    // output never evicts the L2-resident table.
    v4f* __restrict__ op = (v4f*)(out + (size_t)i * (2 * L_LEVELS));
#pragma unroll
    for (int k = 0; k < 8; ++k) {
        v4f v = { acc[4 * k + 0], acc[4 * k + 1], acc[4 * k + 2], acc[4 * k + 3] };
        __builtin_nontemporal_store(v, op + k);
    }
}

extern "C" void kernel_launch(void* const* d_in, const int* in_sizes, int n_in,
                              void* d_out, int out_size, void* d_ws, size_t ws_size,
                              hipStream_t stream)
{
    const float*  xyz   = (const float*)d_in[0];    // (N, 3) f32
    const float2* table = (const float2*)d_in[1];   // (L, T, 2) f32
    float*        out   = (float*)d_out;            // (N, 32) f32

    const int n = in_sizes[0] / 3;                  // 524288
    const int threads = 256;                        // 8 wave32 waves per block
    const int blocks  = (n + threads - 1) / threads;

    hashgrid_encode_kernel<<<blocks, threads, 0, stream>>>(xyz, table, out, n);
}